// RNN_32882269618479
// MI455X (gfx1250) — compile-verified
//
#include <hip/hip_runtime.h>
#include <cmath>

#define SEQ     512
#define BATCH   64
#define IN_DIM  128
#define HDIM    512
#define NWG     8                       // persistent workgroups for the scan
#define COLS_WG (HDIM / NWG)            // 64 output columns per WG
#define LDS_STR 516                     // 512 + 4 pad -> conflict-free banks
#define BH      (BATCH * HDIM)

typedef float v2f __attribute__((ext_vector_type(2)));
typedef float v8f __attribute__((ext_vector_type(8)));

__device__ __forceinline__ v8f wmma4(v2f a, v2f b, v8f c) {
  // D(16x16,f32) = A(16x4,f32) x B(4x16,f32) + C   -- exact f32 math
  return __builtin_amdgcn_wmma_f32_16x16x4_f32(false, a, false, b,
                                               (short)0, c, false, false);
}

// ---------------------------------------------------------------------------
// Kernel 0: zero the global step-barrier counter (replay-deterministic).
// ---------------------------------------------------------------------------
__global__ void rnn_init(unsigned* sync) {
  if (threadIdx.x == 0) *sync = 0u;
}

// ---------------------------------------------------------------------------
// Kernel 1: xi[s,b,h] = sum_i x[s,b,i]*Wi[h,i] + bi[h], written into d_out
// (the h_seq region, consumed/overwritten in place by the scan kernel).
// Block = 256 threads (8 waves). Each block owns one 16-col tile of Wi
// (staged in LDS, reused by all 8 waves); each wave owns one 16-row tile.
// ---------------------------------------------------------------------------
__global__ __launch_bounds__(256) void rnn_proj(const float* __restrict__ x,
                                                const float* __restrict__ Wi,
                                                const float* __restrict__ bi,
                                                float* __restrict__ xi) {
  __shared__ float Wis[16 * 132];           // 16 cols x 128 K, pad to 132
  const int tid    = threadIdx.x;
  const int nTile  = blockIdx.x & 31;       // 32 column tiles of H
  const int mGroup = blockIdx.x >> 5;       // 256 row groups (8 waves each)

  // Stage Wi[nTile*16 .. +16), 128 floats/row, via b128 loads.
  for (int i = tid; i < 16 * 32; i += 256) {
    const int n = i >> 5, k4 = i & 31;
    *(float4*)&Wis[n * 132 + k4 * 4] =
        *(const float4*)&Wi[(size_t)(nTile * 16 + n) * IN_DIM + k4 * 4];
  }
  __syncthreads();

  const int wave = tid >> 5, lane = tid & 31;
  const int l15 = lane & 15, hi = lane >> 4;
  const int mTile = mGroup * 8 + wave;          // 0..2047
  const int rowA  = mTile * 16 + l15;           // A-operand row (per lane)
  const int gcol  = nTile * 16 + l15;           // output column (per lane)

  const float b = bi[gcol];
  v8f c = {b, b, b, b, b, b, b, b};

  const float* xr = x + (size_t)rowA * IN_DIM;
  const float* wr = &Wis[l15 * 132];
#pragma unroll 8
  for (int k = 0; k < IN_DIM; k += 4) {
    const int kq = k + 2 * hi;
    v2f a  = *(const v2f*)&xr[kq];
    v2f bv = *(const v2f*)&wr[kq];
    c = wmma4(a, bv, c);
  }

#pragma unroll
  for (int r = 0; r < 8; ++r) {
    const int row = mTile * 16 + r + 8 * hi;
    xi[(size_t)row * HDIM + gcol] = c[r];
  }
}

// ---------------------------------------------------------------------------
// Kernel 2: persistent recurrence. 8 WGs, each owns 64 columns of H and keeps
// its 512x64 slice of Wh in LDS (132 KB, padded). Per step: read h_{t-1} from
// L2, WMMA-accumulate over K=512, add xi_t (in place) + bh, tanh, store, then
// device-scope monotonic barrier.
// ---------------------------------------------------------------------------
__global__ __launch_bounds__(256) void rnn_scan(const float* __restrict__ Wh,
                                                const float* __restrict__ bh,
                                                float* __restrict__ seq,   // xi in / h_seq out
                                                float* __restrict__ hlast,
                                                unsigned* __restrict__ sync) {
  extern __shared__ float lds[];                 // [COLS_WG][LDS_STR]
  const int tid = threadIdx.x;
  const int colBase = blockIdx.x * COLS_WG;

  // Stage Wh tile: lds[c][k] = Wh[colBase+c][k]  (B-matrix, column-major use)
  for (int i = tid; i < COLS_WG * (HDIM / 4); i += 256) {
    const int c = i >> 7, k4 = i & 127;
    *(float4*)&lds[c * LDS_STR + k4 * 4] =
        *(const float4*)&Wh[(size_t)(colBase + c) * HDIM + k4 * 4];
  }
  __syncthreads();

  const int wave = tid >> 5, lane = tid & 31;
  const int l15 = lane & 15, hi = lane >> 4;
  const int mBase = (wave & 3) * 16;             // 4 row tiles over B=64
  const int nBase = (wave >> 2) * 2;             // 2 col tiles per wave
  const int n0 = nBase * 16 + l15;               // local cols within WG tile
  const int n1 = (nBase + 1) * 16 + l15;
  const int aRow = mBase + l15;                  // batch row for A operand
  const int gcol0 = colBase + n0, gcol1 = colBase + n1;
  const float bh0 = bh[gcol0], bh1 = bh[gcol1];
  const float* b0p = &lds[n0 * LDS_STR];
  const float* b1p = &lds[n1 * LDS_STR];

  for (int t = 0; t < SEQ; ++t) {
    v8f c0 = {}, c1 = {};
    if (t > 0) {
      const float* hp = seq + (size_t)(t - 1) * BH + (size_t)aRow * HDIM;
#pragma unroll 4
      for (int k = 0; k < HDIM; k += 4) {
        const int kq = k + 2 * hi;
        v2f a  = *(const v2f*)&hp[kq];
        v2f bv0 = *(const v2f*)&b0p[kq];
        v2f bv1 = *(const v2f*)&b1p[kq];
        c0 = wmma4(a, bv0, c0);
        c1 = wmma4(a, bv1, c1);
      }
    }

    float* cur = seq + (size_t)t * BH;
#pragma unroll
    for (int r = 0; r < 8; ++r) {
      const int row = mBase + r + 8 * hi;
      const size_t o0 = (size_t)row * HDIM + gcol0;
      const size_t o1 = (size_t)row * HDIM + gcol1;
      const float v0 = tanhf(c0[r] + cur[o0] + bh0);
      const float v1 = tanhf(c1[r] + cur[o1] + bh1);
      cur[o0] = v0;
      cur[o1] = v1;
      if (t == SEQ - 1) { hlast[o0] = v0; hlast[o1] = v1; }
    }

    // -------- device-scope step barrier (monotonic counter) --------
    __builtin_amdgcn_fence(__ATOMIC_RELEASE, "agent");   // flush h_t to L2
    __syncthreads();
    if (tid == 0) {
      __hip_atomic_fetch_add(sync, 1u, __ATOMIC_RELEASE,
                             __HIP_MEMORY_SCOPE_AGENT);
      const unsigned target = (unsigned)NWG * (unsigned)(t + 1);
      while (__hip_atomic_load(sync, __ATOMIC_ACQUIRE,
                               __HIP_MEMORY_SCOPE_AGENT) < target) {
        __builtin_amdgcn_s_sleep(1);
      }
    }
    __syncthreads();
    __builtin_amdgcn_fence(__ATOMIC_ACQUIRE, "agent");   // invalidate WGP$
  }
}

// ---------------------------------------------------------------------------
extern "C" void kernel_launch(void* const* d_in, const int* in_sizes, int n_in,
                              void* d_out, int out_size, void* d_ws, size_t ws_size,
                              hipStream_t stream) {
  const float* x  = (const float*)d_in[0];   // (SEQ,B,IN)
  const float* Wi = (const float*)d_in[1];   // (H,IN)
  const float* bi = (const float*)d_in[2];   // (H)
  const float* Wh = (const float*)d_in[3];   // (H,H)
  const float* bh = (const float*)d_in[4];   // (H)

  float* seq   = (float*)d_out;                          // (SEQ,B,H)
  float* hlast = (float*)d_out + (size_t)SEQ * BH;       // (1,B,H)
  unsigned* sync = (unsigned*)d_ws;

  rnn_init<<<1, 32, 0, stream>>>(sync);

  // (SEQ*B)/(8 waves * 16 rows) row groups x 32 col tiles = 8192 blocks
  rnn_proj<<<dim3((SEQ * BATCH / 128) * 32), dim3(256), 0, stream>>>(x, Wi, bi, seq);

  const size_t lds_bytes = (size_t)COLS_WG * LDS_STR * sizeof(float); // 132 KB
  rnn_scan<<<dim3(NWG), dim3(256), lds_bytes, stream>>>(Wh, bh, seq, hlast, sync);
}